// SACSMAMul_33938831573383
// MI455X (gfx1250) — compile-verified
//
#include <hip/hip_runtime.h>
#include <hip/hip_bf16.h>

#define T_STEPS 365
#define NGRID   2000
#define NMUL    8
#define LENF    15
#define PRECSF  1e-5f

// ---------------------------------------------------------------------------
// CDNA5 async global->LDS copy (gfx1250): (global v4i*, lds v4i*, imm, imm).
// ---------------------------------------------------------------------------
#if defined(__AMDGCN__) && __has_builtin(__builtin_amdgcn_global_load_async_to_lds_b128)
#define HAS_ASYNC_LDS 1
#else
#define HAS_ASYNC_LDS 0
#endif

typedef int v4i_t __attribute__((ext_vector_type(4)));

__device__ __forceinline__ void async_copy16(const float* gsrc, float* ldst) {
#if HAS_ASYNC_LDS
    typedef __attribute__((address_space(1))) v4i_t* gv4_p;
    typedef __attribute__((address_space(3))) v4i_t* lv4_p;
    __builtin_amdgcn_global_load_async_to_lds_b128(
        (gv4_p)gsrc, (lv4_p)ldst, /*offset=*/0, /*cpol=*/0);
#else
    *(float4*)ldst = *(const float4*)gsrc;   // synchronous fallback
#endif
}

template <int N>
__device__ __forceinline__ void wait_async() {
#if HAS_ASYNC_LDS
#if __has_builtin(__builtin_amdgcn_s_wait_asynccnt)
    __builtin_amdgcn_s_wait_asynccnt(N);
#else
    asm volatile("s_wait_asynccnt %0" :: "n"(N) : "memory");
#endif
#endif
}

// ---------------------------------------------------------------------------
// Kernel 1: the sequential SAC-SMA scan.
//   One wave = 4 grid cells x 8 multipliers. Wave-private LDS slices, filled
//   DEPTH timesteps ahead with async global->LDS loads (4 async instructions
//   per batch, in-order completion => a single s_wait_asynccnt gates reuse).
//   No barriers: ASYNCcnt is per-wave and each wave reads only its own slice.
// ---------------------------------------------------------------------------
#define CELLS_PER_WAVE 4
#define SLICE (11 * NMUL * CELLS_PER_WAVE + 4 * CELLS_PER_WAVE)   // 352 + 16 = 368 floats
#define DEPTH 4            // prefetch distance (timesteps)
#define NBUF  (DEPTH + 1)  // LDS buffers per wave

__device__ __forceinline__ void prefetch_slice(const float* __restrict__ params,
                                               const float* __restrict__ x,
                                               int t, int gBase, int lane,
                                               float* slice) {
    // params[t, gBase:gBase+4, 0:11, 0:8] is 352 contiguous floats = 88 x b128
    const float* ps = params + ((size_t)t * NGRID + gBase) * 88;
    async_copy16(ps + (size_t)lane * 4,        slice + lane * 4);
    async_copy16(ps + (size_t)(lane + 32) * 4, slice + (lane + 32) * 4);
    if (lane < 24)
        async_copy16(ps + (size_t)(lane + 64) * 4, slice + (lane + 64) * 4);
    // x[t, gBase:gBase+4, 0:4] -> 4 lanes x b128
    if (lane < 4)
        async_copy16(x + ((size_t)t * NGRID + gBase + lane) * 4,
                     slice + 11 * NMUL * CELLS_PER_WAVE + lane * 4);
}

__global__ __launch_bounds__(256) void sacsma_scan_kernel(
    const float* __restrict__ x,        // [T, NGRID, 4]
    const float* __restrict__ params,   // [T, NGRID, 11, NMUL]
    float* __restrict__ qmean)          // [T, NGRID]
{
    __shared__ float lds[NBUF * 8 * SLICE];

    const int lane  = threadIdx.x & 31;
    const int wave  = threadIdx.x >> 5;
    const int gBase = blockIdx.x * (8 * CELLS_PER_WAVE) + wave * CELLS_PER_WAVE;
    if (gBase >= NGRID) return;                 // uniform per wave (NGRID % 4 == 0)

    const int cell = lane >> 3;                 // 0..3
    const int m    = lane & 7;                  // 0..7
    const int g    = gBase + cell;

    const float LO[11]  = {0.f, 1.f,    0.005f, 0.005f, 0.f, 0.f, 0.005f, 0.005f, 0.f, 0.f, 0.f};
    const float RNG[11] = {1.f, 1999.f, 0.99f,  0.99f,  1.f, 7.f, 0.99f,  0.99f,  1.f, 1.f, 1.f};

    float uztw = 1e-4f, uzfw = 1e-4f, lztw = 1e-4f, lzfwp = 1e-4f, lzfws = 1e-4f;

    // prime the pipeline: t = 0 .. DEPTH-1
#pragma unroll
    for (int d = 0; d < DEPTH; ++d)
        prefetch_slice(params, x, d, gBase, lane, &lds[(d * 8 + wave) * SLICE]);

    for (int t = 0; t < T_STEPS; ++t) {
        if (t + DEPTH < T_STEPS) {
            prefetch_slice(params, x, t + DEPTH, gBase, lane,
                           &lds[(((t + DEPTH) % NBUF) * 8 + wave) * SLICE]);
            wait_async<4 * DEPTH>();  // oldest batch (t) complete, DEPTH remain
        } else {
            wait_async<0>();          // tail: drain (last DEPTH iterations only)
        }

        const float* s = &lds[((t % NBUF) * 8 + wave) * SLICE];

        float pr[11];
#pragma unroll
        for (int p = 0; p < 11; ++p)
            pr[p] = s[cell * 88 + p * 8 + m] * RNG[p] + LO[p];

        const float P  = s[11 * NMUL * CELLS_PER_WAVE + cell * 4 + 0];
        const float Ep = s[11 * NMUL * CELLS_PER_WAVE + cell * 4 + 3];

        const float pctim = pr[0], smax = pr[1], f1 = pr[2], f2 = pr[3], kuz = pr[4];
        const float rexp  = pr[5], f3 = pr[6], f4 = pr[7], pfree = pr[8];
        const float klzp  = pr[9], klzs = pr[10];

        const float uztwm  = fmaxf(PRECSF, f1 * smax);
        const float uzfwm  = fmaxf(PRECSF, f2 * (smax - uztwm));
        const float rem    = smax - uztwm - uzfwm;
        const float lztwm  = fmaxf(PRECSF, f3 * rem);
        const float rem2   = rem - lztwm;
        const float lzfwpm = fmaxf(PRECSF, f4 * rem2);
        const float lzfwsm = fmaxf(PRECSF, (1.f - f4) * rem2);
        const float pbase  = lzfwpm * klzp + lzfwsm * klzs;

        // shared reciprocals (cuts ~half of the f32 division sequences)
        const float i_uztwm = 1.f / uztwm;
        const float i_uzfwm = 1.f / uzfwm;
        const float i_lztwm = 1.f / lztwm;
        const float i_pbase = 1.f / pbase;

        const float zperc = (lztwm + lzfwsm * (1.f - klzs) + lzfwpm * (1.f - klzp)) * i_pbase;

        const float qdir = pctim * P;
        const float peff = (1.f - pctim) * P;

        const float ruz = uztw * i_uztwm;        // uztw / uztwm
        const float rfz = uzfw * i_uzfwm;        // uzfw / uzfwm
        const float ru  = (ruz < rfz)
                        ? (uzfw * uztwm - uztw * uzfwm) / (uztwm + uzfwm) : 0.f;
        const float euztw = fminf(ruz * Ep, uztw);               // dt = 1
        const float twexu = (uztw >= uztwm) ? peff : 0.f;
        const float qsur  = (uzfw >= uzfwm) ? twexu : 0.f;
        const float qint  = kuz * uzfw;
        const float euzfw = fminf(uzfw, fmaxf(Ep - euztw, 0.f));

        const float deficit = fmaxf(lztwm - lztw, 0.f) + fmaxf(lzfwpm - lzfwp, 0.f)
                            + fmaxf(lzfwsm - lzfws, 0.f);
        const float lsum   = lztwm + lzfwpm + lzfwsm;
        const float Srel   = fmaxf(1e-8f, rfz);
        const float demand = pbase * (1.f + zperc * __powf(deficit / lsum, 1.f + rexp));
        const float pc     = fmaxf(1e-8f, fminf(uzfw, Srel * demand));

        const float pctw = (1.f - pfree) * pc;
        const float pcfw = pfree * pc;

        const float rd1 = (lzfwpm - lzfwp) / lzfwpm;
        const float rd2 = (lzfwsm - lzfws) / lzfwsm;
        const float trd = rd1 + rd2;
        const float fsum = lzfwpm + lzfwsm;
        // single divide for both branches of the ternary
        const float flz_num = (trd != 0.f) ? rd1 : lzfwpm;
        const float flz_den = (trd != 0.f) ? trd : fsum;
        const float flz     = flz_num / flz_den;

        const float pcfwp = flz * pcfw;
        const float pcfws = (1.f - flz) * pcfw;

        const float i_fsum = 1.f / fsum;
        const bool  lmask  = (lztw * i_lztwm) < ((lzfwp + lzfws) * i_fsum);
        const float common = lztw * fsum + lztwm * (lzfwp + lzfws);
        const float cd     = common / (fsum * (lztwm + fsum));   // shared by rlp/rls
        const float rlp    = lmask ? (lzfwp * cd) : 0.f;
        const float rls    = lmask ? (lzfws * cd) : 0.f;

        const float elztw  = fminf(lztw * i_lztwm * fmaxf(Ep - euztw - euzfw, 0.f), lztw);
        const float twexl  = (lztw >= lztwm) ? pctw : 0.f;
        const float twexlp = flz * twexl;
        const float twexls = (1.f - flz) * twexl;
        const float qbfp   = klzp * lzfwp;
        const float qbfs   = klzs * lzfws;

        uztw  = fminf(fmaxf(uztw + peff + ru - euztw - twexu, PRECSF), uztwm);
        uzfw  = fminf(fmaxf(uzfw + twexu - ru - euzfw - qint - qsur - pc, PRECSF), uzfwm);
        lztw  = fminf(fmaxf(lztw + pctw - twexl + rlp + rls - elztw, PRECSF), lztwm);
        lzfwp = fminf(fmaxf(lzfwp + pcfwp + twexlp - rlp - qbfp, PRECSF), lzfwpm);
        lzfws = fminf(fmaxf(lzfws + pcfws + twexls - rls - qbfs, PRECSF), lzfwsm);

        float q = qdir + qsur + qint + qbfp + qbfs;

        // mean over the 8 ensemble multipliers (lanes m = 0..7 share a cell)
        q += __shfl_xor(q, 1, 8);
        q += __shfl_xor(q, 2, 8);
        q += __shfl_xor(q, 4, 8);
        if (m == 0) qmean[(size_t)t * NGRID + g] = q * 0.125f;
    }
}

// ---------------------------------------------------------------------------
// Kernel 2: normalized gamma unit-hydrograph weights, w[k, g]
// ---------------------------------------------------------------------------
__global__ void uh_weights_kernel(const float* __restrict__ convp,  // [NGRID, 2]
                                  float* __restrict__ w)            // [LENF, NGRID]
{
    const int g = blockIdx.x * blockDim.x + threadIdx.x;
    if (g >= NGRID) return;
    const float routa = convp[g * 2 + 0] * 2.9f;
    const float routb = convp[g * 2 + 1] * 6.5f;
    const float aa    = fmaxf(routa, 0.f) + 0.1f;
    const float theta = fmaxf(routb, 0.f) + 0.5f;
    const float coef  = __expf(-lgammaf(aa)) * __powf(theta, -aa);
    const float itheta = 1.f / theta;

    float wk[LENF];
    float sum = 0.f;
#pragma unroll
    for (int k = 0; k < LENF; ++k) {
        const float tt = (float)k + 0.5f;
        const float v  = coef * __powf(tt, aa - 1.f) * __expf(-tt * itheta);
        wk[k] = v;
        sum  += v;
    }
    const float inv = 1.f / sum;
#pragma unroll
    for (int k = 0; k < LENF; ++k)
        w[k * NGRID + g] = wk[k] * inv;
}

// ---------------------------------------------------------------------------
// Kernel 3: causal 15-tap convolution, y[t,g] = sum_k w[k,g] * qmean[t-k,g]
//   qmean (2.9 MB) + w (120 KB) are L2-resident (192 MB); coalesced across g.
// ---------------------------------------------------------------------------
__global__ void uh_conv_kernel(const float* __restrict__ qmean,  // [T, NGRID]
                               const float* __restrict__ w,      // [LENF, NGRID]
                               float* __restrict__ out)          // [T, NGRID]
{
    const int idx = blockIdx.x * blockDim.x + threadIdx.x;
    if (idx >= T_STEPS * NGRID) return;
    const int t = idx / NGRID;
    const int g = idx - t * NGRID;

    const int kmax = (t < LENF - 1) ? t : (LENF - 1);
    float acc = 0.f;
    for (int k = 0; k <= kmax; ++k)
        acc = fmaf(w[k * NGRID + g], qmean[(size_t)(t - k) * NGRID + g], acc);
    out[idx] = acc;
}

// ---------------------------------------------------------------------------
extern "C" void kernel_launch(void* const* d_in, const int* in_sizes, int n_in,
                              void* d_out, int out_size, void* d_ws, size_t ws_size,
                              hipStream_t stream) {
    const float* x      = (const float*)d_in[0];  // x_hydro_model [T, NGRID, 4]
    // d_in[1] = c_hydro_model  (unused by the reference forward pass)
    const float* params = (const float*)d_in[2];  // params_raw [T, NGRID, 11, NMUL]
    const float* convp  = (const float*)d_in[3];  // conv_params_hydro [NGRID, 2]

    float* qmean = (float*)d_ws;                             // T*NGRID floats
    float* w     = qmean + (size_t)T_STEPS * NGRID;          // LENF*NGRID floats
    float* out   = (float*)d_out;

    const int blocks1 = (NGRID + 31) / 32;                   // 32 cells per block
    sacsma_scan_kernel<<<blocks1, 256, 0, stream>>>(x, params, qmean);

    uh_weights_kernel<<<(NGRID + 255) / 256, 256, 0, stream>>>(convp, w);

    const int total = T_STEPS * NGRID;
    uh_conv_kernel<<<(total + 255) / 256, 256, 0, stream>>>(qmean, w, out);
}